// fidelity_module_87256555585770
// MI455X (gfx1250) — compile-verified
//
#include <hip/hip_runtime.h>
#include <math.h>

// ---------------- problem constants (from reference) ----------------
#define NPIX  256
#define NANG  20
#define NDET  624
#define NSAMP 768
#define NFFT  2048
#define NBATCH 4

#define F_PI 3.14159265358979323846f

static __device__ __constant__ const float R_SRC  = 59.0f;
static __device__ __constant__ const float D_SD   = 108.0f;   // R_SRC + R_DET

#define XMIN   (-20.0f)
#define DX     (0.15625f)               // 40/256
#define INV_DX (6.4f)                   // 1/DX
#define UMIN   (-60.0f)
#define DUF    ((float)(120.0 / 624.0)) // DU
#define INV_DU ((float)(624.0 / 120.0))
#define UCENT0 ((float)(-60.0 + 0.5 * (120.0 / 624.0)))
#define DTHETA ((float)(2.0 * 3.14159265358979323846 / 20.0))

typedef __attribute__((ext_vector_type(2))) float v2f;
typedef __attribute__((ext_vector_type(8))) float v8f;

// ---------------------------------------------------------------
// Kernel 0: spatial taps of the Hann-windowed ramp filter.
// h[n] = irfft(filt)[n];  filt[m] = (m/(NFFT*DU)) * 0.5*(1+cos(pi*m/1024)),
// filt[0] = filt[1024] = 0, so
// h[n] = (2/NFFT) * sum_{m=1}^{1023} filt[m] * cos(pi*((m*n) mod 2048)/1024)
// ---------------------------------------------------------------
__global__ void k_filter(float* __restrict__ h)
{
    const int n = blockIdx.x * blockDim.x + threadIdx.x;
    if (n >= NFFT) return;
    const float base_freq = (float)(1.0 / (NFFT * (120.0 / 624.0)));
    const float pi_1024   = F_PI / 1024.0f;
    float acc = 0.0f;
    for (int m = 1; m < NFFT / 2; ++m) {
        float freq = (float)m * base_freq;
        float hann = 0.5f * (1.0f + cosf((float)m * pi_1024));
        int   mn   = (m * n) & (NFFT - 1);          // exact modular reduction
        acc += freq * hann * cosf((float)mn * pi_1024);
    }
    h[n] = acc * (2.0f / (float)NFFT);
}

// ---------------------------------------------------------------
// Kernel 1: fan-beam forward projection, one wave32 per ray.
// Fuses: residual (proj - p_tmp) and detector cosine weight -> g[b*A+a][u].
// ---------------------------------------------------------------
__global__ void k_fproj(const float* __restrict__ img,
                        const float* __restrict__ proj,
                        float* __restrict__ g)
{
    const int wave = (blockIdx.x * blockDim.x + threadIdx.x) >> 5;
    const int lane = threadIdx.x & 31;
    if (wave >= NBATCH * NANG * NDET) return;

    const int u  = wave % NDET;
    const int ba = wave / NDET;
    const int a  = ba % NANG;
    const int b  = ba / NANG;

    const float theta = ((float)a + 0.5f) * (2.0f * F_PI / (float)NANG);
    float sn, cs;
    sincosf(theta, &sn, &cs);

    const float uc = UMIN + ((float)u + 0.5f) * DUF;
    const float sx = R_SRC * cs, sy = R_SRC * sn;
    const float px = -49.0f * cs - uc * sn;
    const float py = -49.0f * sn + uc * cs;
    const float rx = px - sx, ry = py - sy;
    const float len = sqrtf(rx * rx + ry * ry);

    const float* __restrict__ im = img + b * NPIX * NPIX;

    float acc = 0.0f;
    for (int s = lane; s < NSAMP; s += 32) {
        float t  = ((float)s + 0.5f) * (1.0f / (float)NSAMP);
        float x  = sx + t * rx;
        float y  = sy + t * ry;
        float gx = (x - XMIN) * INV_DX - 0.5f;
        float gy = (y - XMIN) * INV_DX - 0.5f;
        float i0f = floorf(gx), j0f = floorf(gy);
        float fx = gx - i0f, fy = gy - j0f;
        int i0 = (int)i0f, j0 = (int)j0f;
        bool iv0 = (unsigned)i0       < 256u;
        bool iv1 = (unsigned)(i0 + 1) < 256u;
        bool jv0 = (unsigned)j0       < 256u;
        bool jv1 = (unsigned)(j0 + 1) < 256u;
        float v00 = (iv0 && jv0) ? im[i0 * NPIX + j0]           : 0.0f;
        float v10 = (iv1 && jv0) ? im[(i0 + 1) * NPIX + j0]     : 0.0f;
        float v01 = (iv0 && jv1) ? im[i0 * NPIX + j0 + 1]       : 0.0f;
        float v11 = (iv1 && jv1) ? im[(i0 + 1) * NPIX + j0 + 1] : 0.0f;
        acc += v00 * (1.0f - fx) * (1.0f - fy)
             + v10 * fx          * (1.0f - fy)
             + v01 * (1.0f - fx) * fy
             + v11 * fx          * fy;
    }
    // wave32 reduction
    for (int off = 16; off > 0; off >>= 1)
        acc += __shfl_down(acc, off, 32);

    if (lane == 0) {
        float ptmp = acc * (len * (1.0f / (float)NSAMP));
        float cw   = D_SD / sqrtf(D_SD * D_SD + uc * uc);
        g[wave] = (proj[wave] - ptmp) * cw;
    }
}

// ---------------------------------------------------------------
// Kernel 2: ramp filtering as circulant GEMM on the matrix pipe.
// q[r][j] = DU * sum_{k=0}^{623} g[r][k] * h[(j-k) mod 2048]
// M=80 (5x16 tiles), N=624 (39x16 tiles), K=624 (156 steps of 4).
// One wave32 per 16x16 output tile; chained V_WMMA_F32_16X16X4_F32.
// A layout (f32 16x4): lane -> M = lane%16; K = 2*(lane/16)+v.
// B layout (f32 4x16): lane -> N = lane%16; K = 2*(lane/16)+v.
// D layout (f32 16x16): lane half gives M = r + 8*(lane/16), N = lane%16.
// ---------------------------------------------------------------
__global__ void k_gemm(const float* __restrict__ g,
                       const float* __restrict__ h,
                       float* __restrict__ q)
{
    __shared__ float sh[NFFT];                 // 8 KB of the 320 KB LDS
    const int lane = threadIdx.x;              // blockDim.x == 32
    for (int i = lane; i < NFFT; i += 32)      // uniform trip count, EXEC stays full
        sh[i] = h[i];
    __syncthreads();

    const int tm   = blockIdx.x % 5;           // 5 M-tiles (80 rows)
    const int tn   = blockIdx.x / 5;           // 39 N-tiles (624 cols)
    const int row0 = tm * 16;
    const int col0 = tn * 16;
    const int mn   = lane & 15;                // M for A-frag, N for B-frag
    const int half = lane >> 4;

    v8f c = {};                                // f32 accumulator, 8 VGPRs
    for (int kk = 0; kk < NDET; kk += 4) {
        const float* gp = g + (row0 + mn) * NDET + kk + 2 * half;
        v2f a; a.x = gp[0]; a.y = gp[1];

        int base = col0 + mn - kk - 2 * half;  // circulant index (j - k)
        v2f bm;
        bm.x = sh[ base      & (NFFT - 1)];
        bm.y = sh[(base - 1) & (NFFT - 1)];

        c = __builtin_amdgcn_wmma_f32_16x16x4_f32(
                /*neg_a=*/false, a, /*neg_b=*/false, bm,
                /*c_mod=*/(short)0, c, /*reuse_a=*/false, /*reuse_b=*/false);
    }

#pragma unroll
    for (int r = 0; r < 8; ++r) {
        int mrow = row0 + r + 8 * half;
        q[mrow * NDET + col0 + mn] = c[r] * DUF;
    }
}

// ---------------------------------------------------------------
// Kernel 3: fan-beam filtered back-projection, one thread per pixel.
// ---------------------------------------------------------------
__global__ void k_bproj(const float* __restrict__ q,
                        float* __restrict__ out)
{
    const int idx = blockIdx.x * blockDim.x + threadIdx.x;
    if (idx >= NBATCH * NPIX * NPIX) return;
    const int y = idx & 255;
    const int x = (idx >> 8) & 255;
    const int b = idx >> 16;

    const float X = XMIN + ((float)x + 0.5f) * DX;
    const float Y = XMIN + ((float)y + 0.5f) * DX;

    float acc = 0.0f;
    for (int a = 0; a < NANG; ++a) {
        float theta = ((float)a + 0.5f) * (2.0f * F_PI / (float)NANG);
        float sn, cs;
        sincosf(theta, &sn, &cs);
        float tdist = R_SRC - (X * cs + Y * sn);
        float inv_t = 1.0f / tdist;
        float uu    = D_SD * (-X * sn + Y * cs) * inv_t;
        float k     = (uu - UCENT0) * INV_DU;
        float k0f   = floorf(k);
        float fk    = k - k0f;
        int   k0    = (int)k0f;
        const float* __restrict__ qa = q + (b * NANG + a) * NDET;
        float v0 = ((unsigned)k0       < (unsigned)NDET) ? qa[k0]     : 0.0f;
        float v1 = ((unsigned)(k0 + 1) < (unsigned)NDET) ? qa[k0 + 1] : 0.0f;
        float w  = R_SRC * inv_t;
        w *= w;
        acc += w * (v0 * (1.0f - fk) + v1 * fk);
    }
    out[idx] = 0.5f * DTHETA * acc;
}

// ---------------------------------------------------------------
extern "C" void kernel_launch(void* const* d_in, const int* in_sizes, int n_in,
                              void* d_out, int out_size, void* d_ws, size_t ws_size,
                              hipStream_t stream)
{
    (void)in_sizes; (void)n_in; (void)out_size; (void)ws_size;
    const float* img  = (const float*)d_in[0];   // [4,256,256] f32
    const float* proj = (const float*)d_in[1];   // [4,20,624]  f32
    float* out = (float*)d_out;                  // [4,256,256] f32

    float* ws = (float*)d_ws;
    float* h  = ws;                              // 2048 floats
    float* g  = ws + NFFT;                       // 80*624 floats
    float* q  = ws + NFFT + NBATCH * NANG * NDET;// 80*624 floats

    // 0) spatial filter taps (2048 threads)
    k_filter<<<NFFT / 256, 256, 0, stream>>>(h);

    // 1) forward projection: 49920 rays, one wave32 each, 8 waves/block
    k_fproj<<<(NBATCH * NANG * NDET) / 8, 256, 0, stream>>>(img, proj, g);

    // 2) circulant ramp-filter GEMM: 5*39 = 195 tiles, one wave per tile
    k_gemm<<<5 * 39, 32, 0, stream>>>(g, h, q);

    // 3) back-projection: one thread per output pixel
    k_bproj<<<(NBATCH * NPIX * NPIX) / 256, 256, 0, stream>>>(q, out);
}